// BiRWKVAttention_420906795517
// MI455X (gfx1250) — compile-verified
//
#include <hip/hip_runtime.h>
#include <hip/hip_bf16.h>

// Bidirectional RWKV WKV — chunked two-level decay scan, CDNA5 (gfx1250).
// Memory-bound: ~335MB minimum traffic -> ~14us roofline @ 23.3 TB/s.
// gfx1250 features used:
//   * global_load_async_to_lds_b128 (ASYNCcnt path) to bulk-stage k/v tiles
//   * s_wait_asynccnt
//   * non-temporal TH hints on stream-once r/out so the twice-read k/v
//     stays resident in the 192MB L2 between pass 1 and pass 3.

static constexpr int B   = 4;
static constexpr int T   = 4096;
static constexpr int C   = 1280;
static constexpr int L   = 64;        // chunk length along T
static constexpr int NCH = T / L;     // 64 chunks
static constexpr int TC  = 128;       // channels per block
static constexpr int CT  = C / TC;    // 10 channel tiles

// ---------------------------------------------------------------------------
// gfx1250 async global->LDS helpers
// ---------------------------------------------------------------------------
__device__ __forceinline__ unsigned lds_off_of(const void* p) {
  // Generic pointers into the LDS aperture carry the LDS byte offset in the
  // low 32 bits (CDNA5 ISA).
  return (unsigned)(unsigned long long)p;
}

__device__ __forceinline__ void async_b128(unsigned lds_byte_off, const float* gaddr) {
  // GLOBAL_LOAD_ASYNC_TO_LDS_B128, GV mode: vdst = per-lane LDS byte address,
  // vaddr = per-lane 64-bit global address. Tracked with ASYNCcnt.
  asm volatile("global_load_async_to_lds_b128 %0, %1, off"
               :: "v"(lds_byte_off), "v"(gaddr)
               : "memory");
}

__device__ __forceinline__ void wait_async0() {
#if __has_builtin(__builtin_amdgcn_s_wait_asynccnt)
  __builtin_amdgcn_s_wait_asynccnt(0);
#else
  asm volatile("s_wait_asynccnt 0" ::: "memory");
#endif
}

// Stage a [L][TC] f32 tile (rows t0..t0+L-1, cols c0..c0+TC-1 of a [B,T,C]
// tensor) into LDS with async b128 loads. Each lane moves 16B (4 channels);
// 128 threads cover 4 rows per issue -> L/4 issues per thread per array.
__device__ __forceinline__ void stage_tile(float (*dst)[TC], const float* __restrict__ gsrc,
                                           int b, int t0, int c0, int tid) {
  const int col4   = (tid & 31) * 4;  // 4-channel group within the row
  const int rowoff = tid >> 5;        // which of 4 rows this wave-slot covers
  #pragma unroll
  for (int it = 0; it < L / 4; ++it) {
    const int row = it * 4 + rowoff;
    const float* g = gsrc + ((size_t)b * T + (size_t)(t0 + row)) * C + (c0 + col4);
    async_b128(lds_off_of(&dst[row][col4]), g);
  }
}

// ---------------------------------------------------------------------------
// Pass 1: per-chunk reductions.
//   Xaf = sum_j d^{L-1-j} (e^k v)_j   (forward-weighted)
//   Xbf = sum_j d^{L-1-j} (e^k)_j
//   Xab = sum_j d^{j}     (e^k v)_j   (backward-weighted)
//   Xbb = sum_j d^{j}     (e^k)_j
// ---------------------------------------------------------------------------
__global__ __launch_bounds__(TC) void birwkv_pass1(
    const float* __restrict__ k, const float* __restrict__ v,
    const float* __restrict__ w,
    float* __restrict__ Xaf, float* __restrict__ Xbf,
    float* __restrict__ Xab, float* __restrict__ Xbb) {
  __shared__ float sK[L][TC];
  __shared__ float sV[L][TC];

  const int blk   = blockIdx.x;
  const int ctile = blk % CT;
  const int chunk = (blk / CT) % NCH;
  const int b     = blk / (CT * NCH);
  const int c0    = ctile * TC;
  const int t0    = chunk * L;
  const int tid   = threadIdx.x;
  const int c     = c0 + tid;

  stage_tile(sK, k, b, t0, c0, tid);
  stage_tile(sV, v, b, t0, c0, tid);
  wait_async0();
  __syncthreads();

  const float d = __expf(-__expf(w[c]));
  float af = 0.f, bf = 0.f, ab = 0.f, bb = 0.f, p = 1.f;
  #pragma unroll 4
  for (int t = 0; t < L; ++t) {
    const float ek  = __expf(sK[t][tid]);
    const float ekv = ek * sV[t][tid];
    af = d * af + ekv;   // ends as sum d^{L-1-j} ekv_j
    bf = d * bf + ek;
    ab += p * ekv;       // sum d^{j} ekv_j
    bb += p * ek;
    p *= d;
  }
  const size_t o = ((size_t)b * NCH + chunk) * C + c;
  // Written once, read once (pass 2): keep out of the way of k/v in L2.
  __builtin_nontemporal_store(af, &Xaf[o]);
  __builtin_nontemporal_store(bf, &Xbf[o]);
  __builtin_nontemporal_store(ab, &Xab[o]);
  __builtin_nontemporal_store(bb, &Xbb[o]);
}

// ---------------------------------------------------------------------------
// Pass 2: serial scan over chunk totals -> per-chunk carries.
//   forward carry  S_i:  S_0 = 0,        S_i     = d^L S_{i-1} + Xf[i-1]
//   backward carry R_i:  R_{N-1} = 0,    R_{i-1} = d^L R_i     + Xb[i]
// ---------------------------------------------------------------------------
__global__ void birwkv_pass2(
    const float* __restrict__ w,
    const float* __restrict__ Xaf, const float* __restrict__ Xbf,
    const float* __restrict__ Xab, const float* __restrict__ Xbb,
    float* __restrict__ Saf, float* __restrict__ Sbf,
    float* __restrict__ Rab, float* __restrict__ Rbb) {
  const int idx = blockIdx.x * blockDim.x + threadIdx.x;
  if (idx >= B * C) return;
  const int b = idx / C;
  const int c = idx % C;
  const float dL = __expf(-(float)L * __expf(w[c]));  // d^L = exp(-L e^w)

  float sA = 0.f, sB = 0.f;
  for (int i = 0; i < NCH; ++i) {
    const size_t o = ((size_t)b * NCH + i) * C + c;
    Saf[o] = sA; Sbf[o] = sB;
    sA = dL * sA + __builtin_nontemporal_load(&Xaf[o]);
    sB = dL * sB + __builtin_nontemporal_load(&Xbf[o]);
  }
  float rA = 0.f, rB = 0.f;
  for (int i = NCH - 1; i >= 0; --i) {
    const size_t o = ((size_t)b * NCH + i) * C + c;
    Rab[o] = rA; Rbb[o] = rB;
    rA = dL * rA + __builtin_nontemporal_load(&Xab[o]);
    rB = dL * rB + __builtin_nontemporal_load(&Xbb[o]);
  }
}

// ---------------------------------------------------------------------------
// Pass 3: intra-chunk bidirectional scan + output.
// Backward sweep stores (a_bwd, b_bwd) per timestep in LDS and overwrites the
// staged k/v tiles with e^k and e^k*v (exp computed once). Forward sweep then
// fuses bonus (e^{u+k} = e^u * e^k), receptance gating, and the store.
// ---------------------------------------------------------------------------
__global__ __launch_bounds__(TC) void birwkv_pass3(
    const float* __restrict__ r, const float* __restrict__ k,
    const float* __restrict__ v, const float* __restrict__ w,
    const float* __restrict__ u,
    const float* __restrict__ Saf, const float* __restrict__ Sbf,
    const float* __restrict__ Rab, const float* __restrict__ Rbb,
    float* __restrict__ out) {
  __shared__ float sK [L][TC];  // raw k -> e^k
  __shared__ float sV [L][TC];  // raw v -> e^k * v
  __shared__ float sAB[L][TC];  // a_bwd per timestep
  __shared__ float sBB[L][TC];  // b_bwd per timestep

  const int blk   = blockIdx.x;
  const int ctile = blk % CT;
  const int chunk = (blk / CT) % NCH;
  const int b     = blk / (CT * NCH);
  const int c0    = ctile * TC;
  const int t0    = chunk * L;
  const int tid   = threadIdx.x;
  const int c     = c0 + tid;

  stage_tile(sK, k, b, t0, c0, tid);
  stage_tile(sV, v, b, t0, c0, tid);
  wait_async0();
  __syncthreads();

  const float d  = __expf(-__expf(w[c]));
  const float eu = __expf(u[c]);
  const size_t co = ((size_t)b * NCH + chunk) * C + c;

  // Backward (right-to-left) exclusive scan seeded from the chunk carry.
  float aB = Rab[co], bB = Rbb[co];
  #pragma unroll 4
  for (int t = L - 1; t >= 0; --t) {
    const float ek  = __expf(sK[t][tid]);
    const float ekv = ek * sV[t][tid];
    sK [t][tid] = ek;
    sV [t][tid] = ekv;
    sAB[t][tid] = aB;    // exclusive: state before absorbing x_t
    sBB[t][tid] = bB;
    aB = d * aB + ekv;
    bB = d * bB + ek;
  }
  // No barrier needed: each thread owns column [*][tid].

  // Forward (left-to-right) exclusive scan + fused output.
  float aF = Saf[co], bF = Sbf[co];
  #pragma unroll 4
  for (int t = 0; t < L; ++t) {
    const float ek  = sK[t][tid];
    const float ekv = sV[t][tid];
    const float num = aF + sAB[t][tid] + eu * ekv;   // bonus*v = e^u * e^k v
    const float den = bF + sBB[t][tid] + eu * ek;    // bonus   = e^u * e^k
    const size_t gi = ((size_t)b * T + (size_t)(t0 + t)) * C + c;
    // r is read exactly once and out written exactly once: non-temporal so
    // they don't evict the L2-resident k/v stream.
    const float rr  = __builtin_nontemporal_load(&r[gi]);
    const float sg  = 1.f / (1.f + __expf(-rr));
    __builtin_nontemporal_store(sg * (num / den), &out[gi]);
    aF = d * aF + ekv;
    bF = d * bF + ek;
  }
}

// ---------------------------------------------------------------------------
// Launch
// ---------------------------------------------------------------------------
extern "C" void kernel_launch(void* const* d_in, const int* in_sizes, int n_in,
                              void* d_out, int out_size, void* d_ws, size_t ws_size,
                              hipStream_t stream) {
  const float* r = (const float*)d_in[0];
  const float* k = (const float*)d_in[1];
  const float* v = (const float*)d_in[2];
  const float* w = (const float*)d_in[3];  // [C]
  const float* u = (const float*)d_in[4];  // [C]
  float* out = (float*)d_out;

  // Workspace: 8 arrays of [B, NCH, C] f32 (~10.5 MB).
  float* ws = (float*)d_ws;
  const size_t S = (size_t)B * NCH * C;
  float* Xaf = ws + 0 * S;
  float* Xbf = ws + 1 * S;
  float* Xab = ws + 2 * S;
  float* Xbb = ws + 3 * S;
  float* Saf = ws + 4 * S;
  float* Sbf = ws + 5 * S;
  float* Rab = ws + 6 * S;
  float* Rbb = ws + 7 * S;

  const int nblocks = B * NCH * CT;  // 2560
  birwkv_pass1<<<nblocks, TC, 0, stream>>>(k, v, w, Xaf, Xbf, Xab, Xbb);
  birwkv_pass2<<<(B * C + 255) / 256, 256, 0, stream>>>(w, Xaf, Xbf, Xab, Xbb,
                                                        Saf, Sbf, Rab, Rbb);
  birwkv_pass3<<<nblocks, TC, 0, stream>>>(r, k, v, w, u, Saf, Sbf, Rab, Rbb, out);
}